// Memory_78443282694985
// MI455X (gfx1250) — compile-verified
//
#include <hip/hip_runtime.h>
#include <math.h>

// ---------------- problem constants ----------------
constexpr int kN = 8192;
constexpr int kW = 64;
constexpr int kR = 4;
constexpr float kEPS = 1e-8f;

typedef float f32x2 __attribute__((ext_vector_type(2)));
typedef float f32x4 __attribute__((ext_vector_type(4)));
typedef float f32x8 __attribute__((ext_vector_type(8)));

// ---------------- workspace layout (floats) ----------------
constexpr int WS_SIM  = 0;                  // kN*kR raw dot(memory_row, key_col)
constexpr int WS_CSIM = kN * kR;            // kN   cosine sim vs write key (eps-clamped)
constexpr int WS_WW   = WS_CSIM + kN;       // kN   write weighting
constexpr int WS_FWD  = WS_WW + kN;         // kN*kR  L^T @ rw   (atomically accumulated)
constexpr int WS_BWD  = WS_FWD + kN * kR;   // kN*kR  L   @ rw
constexpr int WS_NRM  = WS_BWD + kN * kR;   // kN   row norms of memory
constexpr int WS_KEYN = WS_NRM + kN;        // kR   read-key column norms
constexpr int WS_WKN  = WS_KEYN + kR;       // 1    write-key norm

// ---------------- output layout (floats, concatenated in return order) ----
constexpr size_t OUT_RV    = 0;                              // (W,R)
constexpr size_t OUT_MEM   = OUT_RV + (size_t)kW * kR;       // (N,W)
constexpr size_t OUT_USAGE = OUT_MEM + (size_t)kN * kW;      // (N)
constexpr size_t OUT_L     = OUT_USAGE + kN;                 // (N,N)
constexpr size_t OUT_P     = OUT_L + (size_t)kN * kN;        // (N)
constexpr size_t OUT_RW    = OUT_P + kN;                     // (N,R)

// ========================================================================
// K0: zero the atomically-accumulated regions; compute key norms.
// ========================================================================
__global__ void k0_init(const float* __restrict__ rkeys,
                        const float* __restrict__ wkey,
                        float* __restrict__ ws, float* __restrict__ out) {
    int gid = blockIdx.x * blockDim.x + threadIdx.x;
    if (gid < 2 * kN * kR) ws[WS_FWD + gid] = 0.0f;          // fwd + bwd (contiguous)
    if (gid < kW * kR)     out[OUT_RV + gid] = 0.0f;         // read_vectors accum
    if (gid < kR) {
        float s = 0.0f;
        for (int k = 0; k < kW; ++k) { float v = rkeys[k * kR + gid]; s += v * v; }
        ws[WS_KEYN + gid] = sqrtf(s);
    }
    if (gid == kR) {
        float s = 0.0f;
        for (int k = 0; k < kW; ++k) { float v = wkey[k]; s += v * v; }
        ws[WS_WKN] = sqrtf(s);
    }
}

// ========================================================================
// K1: WMMA f32 16x16x4 chain: sim = memory(16x64) @ read_keys(64x4, padded
// to 16 cols).  B-operand loads are UNconditional (address clamped with
// lane&3) and padding columns are zeroed with a precomputed multiplicative
// mask -- no divergence, EXEC stays all-ones through the whole WMMA chain.
// A-operand loads are reused for per-row norm^2 and the write-key dot
// (halves combined via shfl_xor(16) per the 16x4 f32 A layout).
// One wave per 16 memory rows.
// ========================================================================
__global__ __launch_bounds__(256) void k1_sim(const float* __restrict__ mem,
                                              const float* __restrict__ rkeys,
                                              const float* __restrict__ wkey,
                                              float* __restrict__ ws) {
    int t = threadIdx.x;
    int lane = t & 31;
    int wv   = t >> 5;
    int half = lane >> 4;      // 0: K pair (0,1) of each 4-chunk, 1: pair (2,3)
    int lrow = lane & 15;      // A: M index   B/D: N (column) index
    int rowBase = (blockIdx.x * (blockDim.x >> 5) + wv) * 16;

    float colmask = (lrow < kR) ? 1.0f : 0.0f;   // zero the 12 padding columns
    int   col     = lrow & 3;                    // clamped (uniform-legal) address

    f32x8 c = {0.f, 0.f, 0.f, 0.f, 0.f, 0.f, 0.f, 0.f};
    float norm2 = 0.0f, wdot = 0.0f;

#pragma unroll
    for (int kb = 0; kb < kW; kb += 4) {
        int k0 = kb + 2 * half;
        f32x2 a = *(const f32x2*)(mem + (size_t)(rowBase + lrow) * kW + k0);
        f32x2 b;
        b.x = rkeys[(size_t)k0 * kR + col] * colmask;
        b.y = rkeys[(size_t)(k0 + 1) * kR + col] * colmask;
        c = __builtin_amdgcn_wmma_f32_16x16x4_f32(false, a, false, b,
                                                  (short)0, c, false, false);
        norm2 += a.x * a.x + a.y * a.y;
        wdot  += a.x * wkey[k0] + a.y * wkey[k0 + 1];
    }
    norm2 += __shfl_xor(norm2, 16);
    wdot  += __shfl_xor(wdot, 16);
    if (half == 0) {
        float nrm = sqrtf(norm2);
        ws[WS_NRM + rowBase + lrow] = nrm;
        float wkn = ws[WS_WKN];
        ws[WS_CSIM + rowBase + lrow] = wdot / fmaxf(nrm * wkn, kEPS);
    }
    // D layout: VGPR v -> row (v + 8*half), col = lane&15. Keep cols 0..R-1.
    if (lrow < kR) {
#pragma unroll
        for (int v = 0; v < 8; ++v) {
            int row = rowBase + v + 8 * half;
            ws[WS_SIM + (size_t)row * kR + lrow] = c[v];
        }
    }
}

// ========================================================================
// K2 (single workgroup, 1024 threads): write-content softmax over N,
// bitonic argsort of usage + exclusive cumprod (allocation weighting,
// including the reference's faithful gather alloc = aw_sorted[order]),
// ww, sum(ww), new_p.   LDS: 32KB + 16KB + 8KB = 56KB.
// ========================================================================
__global__ __launch_bounds__(1024) void k2_alloc(const float* __restrict__ usage,
                                                 const float* __restrict__ prec,
                                                 const float* __restrict__ wstrength,
                                                 const float* __restrict__ agate,
                                                 const float* __restrict__ wgate,
                                                 float* __restrict__ ws,
                                                 float* __restrict__ out) {
    __shared__ float          sv[kN];
    __shared__ unsigned short si[kN];
    __shared__ float          tt[1024];
    __shared__ float          tt2[1024];

    int t = threadIdx.x;
    float wstr = wstrength[0], ag = agate[0], wg = wgate[0];

    // ---- softmax(csim * wstr) over N: max then sum(exp) ----
    float mxl = -3.0e38f;
#pragma unroll
    for (int m = 0; m < 8; ++m)
        mxl = fmaxf(mxl, ws[WS_CSIM + t * 8 + m] * wstr);
    tt[t] = mxl; __syncthreads();
    for (int s = 512; s > 0; s >>= 1) {
        if (t < s) tt[t] = fmaxf(tt[t], tt[t + s]);
        __syncthreads();
    }
    float mx = tt[0]; __syncthreads();
    float sml = 0.0f;
#pragma unroll
    for (int m = 0; m < 8; ++m)
        sml += expf(ws[WS_CSIM + t * 8 + m] * wstr - mx);
    tt[t] = sml; __syncthreads();
    for (int s = 512; s > 0; s >>= 1) {
        if (t < s) tt[t] += tt[t + s];
        __syncthreads();
    }
    float sumexp = tt[0]; __syncthreads();

    // ---- bitonic sort (ascending) of (usage, index) ----
#pragma unroll
    for (int m = 0; m < 8; ++m) {
        int i = t + m * 1024;
        sv[i] = usage[i];
        si[i] = (unsigned short)i;
    }
    __syncthreads();
    for (int k = 2; k <= kN; k <<= 1) {
        for (int j = k >> 1; j > 0; j >>= 1) {
#pragma unroll
            for (int m = 0; m < 8; ++m) {
                int idx = t + m * 1024;
                int ixj = idx ^ j;
                if (ixj > idx) {
                    bool up = ((idx & k) == 0);
                    float va = sv[idx], vb = sv[ixj];
                    if ((va > vb) == up) {
                        sv[idx] = vb; sv[ixj] = va;
                        unsigned short ia = si[idx]; si[idx] = si[ixj]; si[ixj] = ia;
                    }
                }
            }
            __syncthreads();
        }
    }

    // ---- exclusive product scan: register segment + LDS scan of totals ----
    float lv[8], lp[8];
    float prod = 1.0f;
#pragma unroll
    for (int m = 0; m < 8; ++m) {
        lv[m] = sv[t * 8 + m];
        lp[m] = prod;           // exclusive prefix within this 8-segment
        prod *= lv[m];
    }
    tt[t] = prod; __syncthreads();
    float* src = tt; float* dst = tt2;
    for (int off = 1; off < 1024; off <<= 1) {
        float v = src[t];
        if (t >= off) v *= src[t - off];
        dst[t] = v;
        __syncthreads();
        float* tmp = src; src = dst; dst = tmp;
    }
    float base = (t > 0) ? src[t - 1] : 1.0f;
    __syncthreads();

    // aw_sorted -> sv (reuse)
#pragma unroll
    for (int m = 0; m < 8; ++m)
        sv[t * 8 + m] = (1.0f - lv[m]) * base * lp[m];
    __syncthreads();

    // ---- ww = wg*(ag*alloc + (1-ag)*content_w);  alloc[i]=aw_sorted[order[i]]
    float wwl[8];
    float ssum = 0.0f;
#pragma unroll
    for (int m = 0; m < 8; ++m) {
        int i = t * 8 + m;
        float alloc = sv[si[i]];                 // faithful gather quirk
        float cw = expf(ws[WS_CSIM + i] * wstr - mx) / sumexp;
        float v = wg * (ag * alloc + (1.0f - ag) * cw);
        wwl[m] = v;
        ws[WS_WW + i] = v;
        ssum += v;
    }
    tt[t] = ssum; __syncthreads();
    for (int s = 512; s > 0; s >>= 1) {
        if (t < s) tt[t] += tt[t + s];
        __syncthreads();
    }
    float om = 1.0f - tt[0];
#pragma unroll
    for (int m = 0; m < 8; ++m) {
        int i = t * 8 + m;
        out[OUT_P + i] = om * prec[i] + wwl[m];
    }
}

// ========================================================================
// K3: THE roofline kernel. One fused streaming pass over L (256MB read,
// 256MB NT write):  new_L = (1-ww_j-ww_i)*L + ww_i*p_j  while accumulating
// fwd[j,r] += L[i,j]*rw[i,r]  and  bwd[i,r] += L[i,j]*rw[j,r].
// Tile 64(i) x 128(j) per 256-thread block; lane owns 4 j-cols, wave owns
// 8 i-rows.  bwd reduced wave-locally via shfl_xor; fwd via LDS ds_add_f32
// then one global atomic per (j,r).
// ========================================================================
constexpr int BI = 64;
constexpr int BJ = 128;

__global__ __launch_bounds__(256) void k3_linkage(const float* __restrict__ L,
                                                  const float* __restrict__ rw,
                                                  const float* __restrict__ prec,
                                                  float* __restrict__ ws,
                                                  float* __restrict__ out) {
    __shared__ float fwdL[BJ * kR];
    int t = threadIdx.x;
    int lane = t & 31, wv = t >> 5;
    int j0 = blockIdx.x * BJ;
    int i0 = blockIdx.y * BI;
    const float* ww = ws + WS_WW;

    int jb = j0 + lane * 4;
    f32x4 wwj = *(const f32x4*)(ww + jb);
    f32x4 pj  = *(const f32x4*)(prec + jb);
    f32x4 rwj[4];
#pragma unroll
    for (int jj = 0; jj < 4; ++jj)
        rwj[jj] = *(const f32x4*)(rw + (size_t)(jb + jj) * kR);

    float wwi[8];
    f32x4 rwi[8];
#pragma unroll
    for (int k = 0; k < 8; ++k) {
        int i = i0 + wv + 8 * k;
        wwi[k] = ww[i];
        rwi[k] = *(const f32x4*)(rw + (size_t)i * kR);
    }

    f32x4 fwd_acc[4] = {};   // [jj] -> over r
    f32x4 bwd_acc[8] = {};   // [k]  -> over r
    float* outL = out + OUT_L;

#pragma unroll
    for (int k = 0; k < 8; ++k) {
        int i = i0 + wv + 8 * k;
        size_t off = (size_t)i * kN + jb;
        f32x4 e = __builtin_nontemporal_load((const f32x4*)(L + off));
        f32x4 nl;
#pragma unroll
        for (int jj = 0; jj < 4; ++jj) {
            float lv = e[jj];
            nl[jj] = (1.0f - wwj[jj] - wwi[k]) * lv + wwi[k] * pj[jj];
            fwd_acc[jj] += lv * rwi[k];
            bwd_acc[k]  += lv * rwj[jj];
        }
        __builtin_nontemporal_store(nl, (f32x4*)(outL + off));
    }

    // bwd: all lanes of a wave share the same i rows -> wave reduction
#pragma unroll
    for (int k = 0; k < 8; ++k) {
#pragma unroll
        for (int r = 0; r < kR; ++r) {
            float v = bwd_acc[k][r];
            for (int off = 16; off > 0; off >>= 1) v += __shfl_xor(v, off);
            if (lane == 0)
                atomicAdd(ws + WS_BWD + (size_t)(i0 + wv + 8 * k) * kR + r, v);
        }
    }

    // fwd: reduce across the 8 waves via LDS, then one global atomic each
    for (int x = t; x < BJ * kR; x += 256) fwdL[x] = 0.0f;
    __syncthreads();
#pragma unroll
    for (int jj = 0; jj < 4; ++jj)
#pragma unroll
        for (int r = 0; r < kR; ++r)
            atomicAdd(&fwdL[(lane * 4 + jj) * kR + r], fwd_acc[jj][r]);
    __syncthreads();
    for (int x = t; x < BJ * kR; x += 256)
        atomicAdd(ws + WS_FWD + (size_t)j0 * kR + x, fwdL[x]);
}

// ========================================================================
// K4: per-row read weighting, retention, new usage.
// ========================================================================
__global__ void k4_readw(const float* __restrict__ usage,
                         const float* __restrict__ beta,
                         const float* __restrict__ rmw,
                         const float* __restrict__ fgate,
                         float* __restrict__ ws, float* __restrict__ out) {
    int i = blockIdx.x * blockDim.x + threadIdx.x;
    f32x4 s    = *(const f32x4*)(ws + WS_SIM + (size_t)i * kR);
    float nrm  = ws[WS_NRM + i];
    f32x4 keyn = *(const f32x4*)(ws + WS_KEYN);
    f32x4 bt   = *(const f32x4*)beta;
    f32x4 rm0  = *(const f32x4*)(rmw);
    f32x4 rm1  = *(const f32x4*)(rmw + kR);
    f32x4 rm2  = *(const f32x4*)(rmw + 2 * kR);
    f32x4 fg   = *(const f32x4*)fgate;
    f32x4 fwd  = *(const f32x4*)(ws + WS_FWD + (size_t)i * kR);
    f32x4 bwd  = *(const f32x4*)(ws + WS_BWD + (size_t)i * kR);
    float wwi  = ws[WS_WW + i];
    float u    = usage[i];

    f32x4 x;
#pragma unroll
    for (int r = 0; r < kR; ++r) x[r] = (s[r] / (nrm * keyn[r])) * bt[r];
    float m = fmaxf(fmaxf(x[0], x[1]), fmaxf(x[2], x[3]));
    f32x4 e;
#pragma unroll
    for (int r = 0; r < kR; ++r) e[r] = expf(x[r] - m);
    float es = e[0] + e[1] + e[2] + e[3];

    f32x4 rdw;
    float ret = 1.0f;
#pragma unroll
    for (int r = 0; r < kR; ++r) {
        rdw[r] = rm0[r] * bwd[r] + rm1[r] * (e[r] / es) + rm2[r] * fwd[r];
        ret *= (1.0f - rdw[r] * fg[r]);
    }
    *(f32x4*)(out + OUT_RW + (size_t)i * kR) = rdw;
    out[OUT_USAGE + i] = (u + wwi - u * wwi) * ret;
}

// ========================================================================
// K5: new_memory = memory * (1 - ww_i * erase_w) + ww_i * write_vector_w
// ========================================================================
__global__ void k5_mem(const float* __restrict__ mem,
                       const float* __restrict__ erase,
                       const float* __restrict__ wvec,
                       const float* __restrict__ ws, float* __restrict__ out) {
    int idx4 = blockIdx.x * blockDim.x + threadIdx.x;   // one float4 each
    int i  = idx4 >> 4;
    int w4 = idx4 & 15;
    f32x4 m  = *(const f32x4*)(mem + (size_t)i * kW + w4 * 4);
    f32x4 er = *(const f32x4*)(erase + w4 * 4);
    f32x4 wv = *(const f32x4*)(wvec + w4 * 4);
    float wwi = ws[WS_WW + i];
    f32x4 o = m * (1.0f - wwi * er) + wwi * wv;
    *(f32x4*)(out + OUT_MEM + (size_t)idx4 * 4) = o;
}

// ========================================================================
// K6: read_vectors = memory^T @ read_w  (64x8192 @ 8192x4 -> 64x4)
// ========================================================================
__global__ __launch_bounds__(256) void k6_rv(const float* __restrict__ mem,
                                             const float* __restrict__ rdw,
                                             float* __restrict__ out) {
    int t = threadIdx.x;
    int w4 = t & 15, rg = t >> 4;
    f32x4 acc[4] = {};
    int base = blockIdx.x * 1024;
#pragma unroll 4
    for (int k = 0; k < 64; ++k) {
        int i = base + rg + 16 * k;
        f32x4 m  = *(const f32x4*)(mem + (size_t)i * kW + w4 * 4);
        f32x4 r4 = *(const f32x4*)(rdw + (size_t)i * kR);
#pragma unroll
        for (int wj = 0; wj < 4; ++wj) acc[wj] += m[wj] * r4;
    }
#pragma unroll
    for (int wj = 0; wj < 4; ++wj)
#pragma unroll
        for (int r = 0; r < kR; ++r)
            atomicAdd(out + OUT_RV + (size_t)(w4 * 4 + wj) * kR + r, acc[wj][r]);
}

// ========================================================================
extern "C" void kernel_launch(void* const* d_in, const int* in_sizes, int n_in,
                              void* d_out, int out_size, void* d_ws, size_t ws_size,
                              hipStream_t stream) {
    const float* mem   = (const float*)d_in[0];
    const float* usage = (const float*)d_in[1];
    const float* prec  = (const float*)d_in[2];
    const float* L     = (const float*)d_in[3];
    const float* rw    = (const float*)d_in[4];
    const float* rkeys = (const float*)d_in[5];
    const float* beta  = (const float*)d_in[6];
    const float* rmw   = (const float*)d_in[7];
    const float* wkey  = (const float*)d_in[8];
    const float* wstr  = (const float*)d_in[9];
    const float* ag    = (const float*)d_in[10];
    const float* wg    = (const float*)d_in[11];
    const float* er    = (const float*)d_in[12];
    const float* wv    = (const float*)d_in[13];
    const float* fg    = (const float*)d_in[14];
    float* out = (float*)d_out;
    float* ws  = (float*)d_ws;

    k0_init <<<257, 256, 0, stream>>>(rkeys, wkey, ws, out);
    k1_sim  <<<kN / 128, 256, 0, stream>>>(mem, rkeys, wkey, ws);
    k2_alloc<<<1, 1024, 0, stream>>>(usage, prec, wstr, ag, wg, ws, out);
    k3_linkage<<<dim3(kN / BJ, kN / BI), 256, 0, stream>>>(L, rw, prec, ws, out);
    k4_readw<<<kN / 256, 256, 0, stream>>>(usage, beta, rmw, fg, ws, out);
    k5_mem  <<<(kN * kW / 4) / 256, 256, 0, stream>>>(mem, er, wv, ws, out);
    k6_rv   <<<kN / 1024, 256, 0, stream>>>(mem, out + OUT_RW, out);
}